// CWTScalogramLayer_2216203125161
// MI455X (gfx1250) — compile-verified
//
#include <hip/hip_runtime.h>
#include <hip/hip_bf16.h>
#include <math.h>

typedef __attribute__((ext_vector_type(2))) float v2f;
typedef __attribute__((ext_vector_type(8))) float v8f;

#define SCALES 64
#define NB     128
#define NLEN   2048
#define IMGSZ  224

#ifdef __has_builtin
#if __has_builtin(__builtin_amdgcn_wmma_f32_16x16x4_f32)
#define HAVE_WMMA_F32F32 1
#endif
#endif
#ifndef HAVE_WMMA_F32F32
#define HAVE_WMMA_F32F32 0
#endif

__device__ __forceinline__ v8f wmma_f32(v2f a, v2f b, v8f c) {
#if HAVE_WMMA_F32F32
  return __builtin_amdgcn_wmma_f32_16x16x4_f32(false, a, false, b,
                                               (short)0, c, false, false);
#else
  c[0] += a[0] * b[0] + a[1] * b[1];   // compile-safe placeholder
  return c;
#endif
}

// ---------------- Stage A: int_psi table (cumsum of Morlet, fp64) ----------
__global__ void build_int_psi(float* __restrict__ int_psi) {
  __shared__ double psi[1024];
  int i = threadIdx.x;
  double x = -8.0 + 16.0 * (double)i / 1023.0;
  psi[i] = exp(-x * x * 0.5) * cos(5.0 * x);
  __syncthreads();
  if (i == 0) {
    const double step = 16.0 / 1023.0;
    double acc = 0.0;
    for (int t = 0; t < 1024; ++t) {
      acc += psi[t];
      int_psi[t] = (float)(acc * step);
    }
  }
}

// ---------------- Stage B: per-scale diff'd kernels w_s[j] = -sqrt(s)*(k[j-1]-k[j])
__global__ void build_weights(const float* __restrict__ int_psi,
                              float* __restrict__ w_all) {
  int s  = blockIdx.x + 1;
  int K  = 16 * s + 1;
  int Kw = K + 1;
  int woff = (s - 1) * (8 * s + 2);
  float sq = sqrtf((float)s);
  double sstep = (double)s * (16.0 / 1023.0);
  for (int j = threadIdx.x; j < Kw; j += blockDim.x) {
    float km1 = 0.f, k0 = 0.f;
    int t = j - 1;
    if (t >= 0 && t < K) {
      int id = (int)((double)t / sstep);
      if (id > 1023) id = 1023;
      km1 = int_psi[id];
    }
    if (j < K) {
      int id = (int)((double)j / sstep);
      if (id > 1023) id = 1023;
      k0 = int_psi[id];
    }
    w_all[woff + j] = -sq * (km1 - k0);
  }
}

// ---------------- Stage C: sliding correlation as Toeplitz matmul via WMMA ----
// coef[b, n] = sum_j w_s[j] * x[b, n + j - 8s - 1];  scalo = |coef|
// Block: one scale x 32-batch pair x 128 columns.
// Wave: two 16x16 output tiles (batch-paired) sharing one B (weights) fragment.
template <bool CHECKED>
__device__ __forceinline__ void conv_chunk(int c, int base, int kk, int m, int Kw,
                                           const float* __restrict__ xrow0,
                                           const float* __restrict__ xrow1,
                                           const float* __restrict__ wsh,
                                           v8f& acc0, v8f& acc1) {
  int u   = (c << 2) + kk;
  int col = base + u;
  int t0  = u - m;
  float ax0, ay0, ax1, ay1, bx, by;
  if (CHECKED) {
    bool i0 = (col >= 0) && (col < NLEN);
    bool i1 = (col + 1 >= 0) && (col + 1 < NLEN);
    ax0 = i0 ? xrow0[col] : 0.f;
    ay0 = i1 ? xrow0[col + 1] : 0.f;
    ax1 = i0 ? xrow1[col] : 0.f;
    ay1 = i1 ? xrow1[col + 1] : 0.f;
    bx = (t0 >= 0 && t0 < Kw) ? wsh[t0] : 0.f;
    by = (t0 + 1 >= 0 && t0 + 1 < Kw) ? wsh[t0 + 1] : 0.f;
  } else {
    ax0 = xrow0[col];
    ay0 = xrow0[col + 1];
    ax1 = xrow1[col];
    ay1 = xrow1[col + 1];
    bx = wsh[t0];
    by = wsh[t0 + 1];
  }
  v2f a0 = {ax0, ay0};
  v2f a1 = {ax1, ay1};
  v2f b  = {bx, by};
  acc0 = wmma_f32(a0, b, acc0);
  acc1 = wmma_f32(a1, b, acc1);
}

__global__ void __launch_bounds__(256) cwt_conv_wmma(const float* __restrict__ x,
                                                     const float* __restrict__ w_all,
                                                     float* __restrict__ scalo) {
  __shared__ float wsh[1026];                 // max Kw = 16*64+2
  int bid   = blockIdx.x;
  int s_idx = bid >> 6;                       // 0..63
  int rem   = bid & 63;
  int bt    = rem >> 4;                       // batch pair tile 0..3 (32 batches)
  int ng    = rem & 15;                       // column group 0..15
  int s     = s_idx + 1;
  int Kw    = 16 * s + 2;
  int woff  = (s - 1) * (8 * s + 2);
  for (int j = threadIdx.x; j < Kw; j += 256) wsh[j] = w_all[woff + j];
  __syncthreads();

  int wave = threadIdx.x >> 5;
  int lane = threadIdx.x & 31;
  int m    = lane & 15;                       // row (A) / column (B,D) index
  int half = lane >> 4;
  int kk   = half << 1;                       // K sub-index 0 or 2
  int n0   = ng * 128 + wave * 16;
  int b0   = bt * 32;
  int base = n0 - 8 * s - 1;

  const float* xrow0 = x + (size_t)(b0 + m) * NLEN;
  const float* xrow1 = xrow0 + (size_t)16 * NLEN;
  v8f acc0 = {0.f, 0.f, 0.f, 0.f, 0.f, 0.f, 0.f, 0.f};
  v8f acc1 = {0.f, 0.f, 0.f, 0.f, 0.f, 0.f, 0.f, 0.f};
  int chunks = 4 * s + 5;                     // = (Kw + 15 + 3) / 4

  // Wave-uniform interior range: no lane touches x or weight band out of range.
  int cIntLo = 4;                                       // t0 >= 0 for all lanes
  if (base < 0) {
    int t = ((-base) + 3) >> 2;                         // col >= 0 for all lanes
    if (t > cIntLo) cIntLo = t;
  }
  int cIntHi = (Kw - 4) >> 2;                           // t0+1 < Kw for all lanes
  {
    int t = (NLEN - 5 - base) >> 2;                     // col+1 < NLEN for all lanes
    if (t < cIntHi) cIntHi = t;
  }
  if (cIntHi > chunks - 1) cIntHi = chunks - 1;
  int cEdge0 = (cIntLo < chunks) ? cIntLo : chunks;

  int c = 0;
  for (; c < cEdge0; ++c)
    conv_chunk<true>(c, base, kk, m, Kw, xrow0, xrow1, wsh, acc0, acc1);
  for (; c <= cIntHi; ++c)
    conv_chunk<false>(c, base, kk, m, Kw, xrow0, xrow1, wsh, acc0, acc1);
  for (; c < chunks; ++c)
    conv_chunk<true>(c, base, kk, m, Kw, xrow0, xrow1, wsh, acc0, acc1);

  // D layout: VGPR r -> M = r + half*8, N = m
  for (int r = 0; r < 8; ++r) {
    int bb0 = b0 + r + half * 8;
    scalo[((size_t)bb0 * SCALES + s_idx) * NLEN + n0 + m] = fabsf(acc0[r]);
    scalo[((size_t)(bb0 + 16) * SCALES + s_idx) * NLEN + n0 + m] = fabsf(acc1[r]);
  }
}

// ---------------- Cubic (Keys, a=-0.5) kernel ------------------------------
__device__ __forceinline__ float keys_cubic(float t) {
  t = fabsf(t);
  float t2 = t * t;
  float t3 = t2 * t;
  if (t <= 1.f) return 1.5f * t3 - 2.5f * t2 + 1.f;
  if (t <  2.f) return -0.5f * t3 + 2.5f * t2 - 4.f * t + 2.f;
  return 0.f;
}

// ---------------- Stage D1: width resize 2048 -> 224 (antialiased downsample)
__global__ void resize_w(const float* __restrict__ scalo, float* __restrict__ tmp1) {
  int idx = blockIdx.x * blockDim.x + threadIdx.x;
  const int total = NB * SCALES * IMGSZ;
  if (idx >= total) return;
  int ow = idx % IMGSZ;
  int rest = idx / IMGSZ;
  int s = rest % SCALES;
  int b = rest / SCALES;
  const float scale = (float)IMGSZ / (float)NLEN;       // < 1: downsample
  float center = (ow + 0.5f) / scale - 0.5f;
  float radius = 2.f / scale;
  int j0 = (int)ceilf(center - radius);
  int j1 = (int)floorf(center + radius);
  if (j0 < 0) j0 = 0;
  if (j1 > NLEN - 1) j1 = NLEN - 1;
  const float* row = scalo + ((size_t)b * SCALES + s) * NLEN;
  float av = 0.f, aw = 0.f;
  for (int j = j0; j <= j1; ++j) {
    float wgt = keys_cubic((j - center) * scale);
    aw += wgt;
    av += wgt * row[j];
  }
  tmp1[idx] = av / aw;
}

// ---------------- Stage D2: height resize 64 -> 224 (upsample) --------------
__global__ void resize_h(const float* __restrict__ tmp1, float* __restrict__ img) {
  int idx = blockIdx.x * blockDim.x + threadIdx.x;
  const int total = NB * IMGSZ * IMGSZ;
  if (idx >= total) return;
  int ow = idx % IMGSZ;
  int rest = idx / IMGSZ;
  int oh = rest % IMGSZ;
  int b = rest / IMGSZ;
  const float scale = (float)IMGSZ / (float)SCALES;      // 3.5: upsample
  float center = (oh + 0.5f) / scale - 0.5f;
  int j0 = (int)ceilf(center - 2.f);
  int j1 = (int)floorf(center + 2.f);
  if (j0 < 0) j0 = 0;
  if (j1 > SCALES - 1) j1 = SCALES - 1;
  float av = 0.f, aw = 0.f;
  for (int j = j0; j <= j1; ++j) {
    float wgt = keys_cubic(j - center);
    aw += wgt;
    av += wgt * tmp1[((size_t)b * SCALES + j) * IMGSZ + ow];
  }
  img[idx] = av / aw;
}

// ---------------- Stage E: per-image min/max --------------------------------
__global__ void __launch_bounds__(256) minmax_reduce(const float* __restrict__ img,
                                                     float* __restrict__ mnb,
                                                     float* __restrict__ mxb) {
  __shared__ float smn[256];
  __shared__ float smx[256];
  int b = blockIdx.x;
  const float* base = img + (size_t)b * IMGSZ * IMGSZ;
  float mn = 3.4e38f, mx = -3.4e38f;
  for (int i = threadIdx.x; i < IMGSZ * IMGSZ; i += 256) {
    float v = base[i];
    mn = fminf(mn, v);
    mx = fmaxf(mx, v);
  }
  smn[threadIdx.x] = mn;
  smx[threadIdx.x] = mx;
  __syncthreads();
  for (int off = 128; off > 0; off >>= 1) {
    if (threadIdx.x < off) {
      smn[threadIdx.x] = fminf(smn[threadIdx.x], smn[threadIdx.x + off]);
      smx[threadIdx.x] = fmaxf(smx[threadIdx.x], smx[threadIdx.x + off]);
    }
    __syncthreads();
  }
  if (threadIdx.x == 0) {
    mnb[b] = smn[0];
    mxb[b] = smx[0];
  }
}

// ---------------- Stage F: normalize + channel broadcast + mean/std ---------
__global__ void finalize(const float* __restrict__ img,
                         const float* __restrict__ mnb,
                         const float* __restrict__ mxb,
                         const float* __restrict__ mean,
                         const float* __restrict__ stdv,
                         float* __restrict__ out) {
  int idx = blockIdx.x * blockDim.x + threadIdx.x;
  const int total = NB * 3 * IMGSZ * IMGSZ;
  if (idx >= total) return;
  int w = idx % IMGSZ;
  int h = (idx / IMGSZ) % IMGSZ;
  int c = (idx / (IMGSZ * IMGSZ)) % 3;
  int b = idx / (3 * IMGSZ * IMGSZ);
  float v = img[((size_t)b * IMGSZ + h) * IMGSZ + w];
  float mn = mnb[b], mx = mxb[b];
  float nv = (v - mn) / (mx - mn + 1e-8f);
  out[idx] = (nv - mean[c]) / stdv[c];
}

extern "C" void kernel_launch(void* const* d_in, const int* in_sizes, int n_in,
                              void* d_out, int out_size, void* d_ws, size_t ws_size,
                              hipStream_t stream) {
  const float* x    = (const float*)d_in[0];   // (128, 2048)
  const float* mean = (const float*)d_in[1];   // 3
  const float* stdv = (const float*)d_in[2];   // 3
  float* out = (float*)d_out;                  // (128, 3, 224, 224)

  float* ws      = (float*)d_ws;
  float* int_psi = ws;                                   // 1024
  float* w_all   = int_psi + 1024;                       // 33408
  float* scalo   = w_all + 33408;                        // 128*64*2048
  float* tmp1    = scalo + (size_t)NB * SCALES * NLEN;   // 128*64*224
  float* img     = tmp1 + (size_t)NB * SCALES * IMGSZ;   // 128*224*224
  float* mnb     = img + (size_t)NB * IMGSZ * IMGSZ;     // 128
  float* mxb     = mnb + NB;                             // 128

  build_int_psi<<<1, 1024, 0, stream>>>(int_psi);
  build_weights<<<SCALES, 256, 0, stream>>>(int_psi, w_all);

  // 64 scales * 4 batch-pair tiles * 16 column-groups
  cwt_conv_wmma<<<SCALES * 4 * 16, 256, 0, stream>>>(x, w_all, scalo);

  {
    int total = NB * SCALES * IMGSZ;
    resize_w<<<(total + 255) / 256, 256, 0, stream>>>(scalo, tmp1);
  }
  {
    int total = NB * IMGSZ * IMGSZ;
    resize_h<<<(total + 255) / 256, 256, 0, stream>>>(tmp1, img);
  }
  minmax_reduce<<<NB, 256, 0, stream>>>(img, mnb, mxb);
  {
    int total = NB * 3 * IMGSZ * IMGSZ;
    finalize<<<(total + 255) / 256, 256, 0, stream>>>(img, mnb, mxb, mean, stdv, out);
  }
}